// Mass_conservation_loss_13082470383973
// MI455X (gfx1250) — compile-verified
//
#include <hip/hip_runtime.h>

typedef float v2f  __attribute__((ext_vector_type(2)));
typedef float v8f  __attribute__((ext_vector_type(8)));
typedef float f32x4 __attribute__((ext_vector_type(4)));

#define NMAT 8192
#define TILE_COLS 1024          // columns per block
#define ROWS_PER_BLOCK 64       // rows per block
#define RCHUNK 16               // rows staged in LDS at a time (matches WMMA M)
#define LDS_STRIDE (TILE_COLS + 4)  // pad: row r starts at bank 4r -> conflict-free A-frag reads

// ---------------------------------------------------------------------------
// Kernel 1: one pass over flow; row sums via V_WMMA_F32_16X16X4_F32 against a
// ones B-matrix, column sums via per-thread register accumulation at load time.
// ---------------------------------------------------------------------------
__global__ __launch_bounds__(256) void sums_kernel(const float* __restrict__ flow,
                                                   float* __restrict__ row_sums,
                                                   float* __restrict__ col_sums) {
    __shared__ float tile[RCHUNK * LDS_STRIDE];   // 16 x 1028 f32 = 64.25 KB

    const int tid  = threadIdx.x;
    const int lane = tid & 31;
    const int wave = tid >> 5;                    // 8 waves
    const int col0 = blockIdx.x * TILE_COLS;
    const int row0 = blockIdx.y * ROWS_PER_BLOCK;
    const int c4   = tid * 4;                     // this thread's 4 columns in tile

    f32x4 col_acc = {0.f, 0.f, 0.f, 0.f};

    // A-fragment addressing for 32-bit 16x4 A (ISA 7.12.2):
    // lane L holds row M = L&15; VGPR0/1 = (M, k+koff), (M, k+koff+1), koff = (L>=16)?2:0
    const int m    = lane & 15;
    const int koff = (lane >> 4) << 1;
    const v2f bones = {1.0f, 1.0f};               // 4x16 all-ones B fragment (layout-invariant)

    for (int rc = 0; rc < ROWS_PER_BLOCK / RCHUNK; ++rc) {
        // --- stage 16 rows x 1024 cols into LDS; coalesced b128 global loads ---
        for (int r = 0; r < RCHUNK; ++r) {
            const float* src = flow + (size_t)(row0 + rc * RCHUNK + r) * NMAT + (col0 + c4);
            f32x4 v = *(const f32x4*)src;
            col_acc += v;                                         // column partials (free)
            *(f32x4*)&tile[r * LDS_STRIDE + c4] = v;              // b128 LDS store (1028%4==0)
        }
        __syncthreads();

        // --- row sums: wave w reduces its 128-column segment with 32 WMMAs ---
        v8f c = {0.f, 0.f, 0.f, 0.f, 0.f, 0.f, 0.f, 0.f};
        const int kbase = wave * 128;
        #pragma unroll 4
        for (int k = 0; k < 128; k += 4) {
            const float* p = &tile[m * LDS_STRIDE + kbase + k + koff];
            v2f a = { p[0], p[1] };
            // D = A(16x4) * Ones(4x16) + C : every column of D = row sum of A
            c = __builtin_amdgcn_wmma_f32_16x16x4_f32(false, a, false, bones,
                                                      (short)0, c, false, false);
        }

        // --- extract: C VGPR i holds M=i (lanes 0-15) and M=8+i (lanes 16-31) ---
        const int rbase = row0 + rc * RCHUNK;
        if (lane == 0) {
            #pragma unroll
            for (int i = 0; i < 8; ++i) unsafeAtomicAdd(&row_sums[rbase + i], c[i]);
        } else if (lane == 16) {
            #pragma unroll
            for (int i = 0; i < 8; ++i) unsafeAtomicAdd(&row_sums[rbase + 8 + i], c[i]);
        }
        __syncthreads();   // reconverge (EXEC all-1s for next WMMA loop) + LDS reuse
    }

    // --- flush column partials (64 rows worth) ---
    #pragma unroll
    for (int i = 0; i < 4; ++i) unsafeAtomicAdd(&col_sums[col0 + c4 + i], col_acc[i]);
}

// ---------------------------------------------------------------------------
// Kernel 0: zero the accumulators (must run every launch: atomics accumulate).
// ---------------------------------------------------------------------------
__global__ void zero_kernel(float* p, int n) {
    int i = blockIdx.x * blockDim.x + threadIdx.x;
    if (i < n) p[i] = 0.f;
}

// ---------------------------------------------------------------------------
// Kernel 2: out = sum_i |row_sums[i] - col_sums[i]|   (8192 elements, 1 block)
// ---------------------------------------------------------------------------
__global__ __launch_bounds__(256) void finalize_kernel(const float* __restrict__ row_sums,
                                                       const float* __restrict__ col_sums,
                                                       float* __restrict__ out) {
    __shared__ float red[256];
    float acc = 0.f;
    for (int i = threadIdx.x; i < NMAT; i += 256)
        acc += fabsf(row_sums[i] - col_sums[i]);
    red[threadIdx.x] = acc;
    __syncthreads();
    for (int s = 128; s > 0; s >>= 1) {
        if (threadIdx.x < s) red[threadIdx.x] += red[threadIdx.x + s];
        __syncthreads();
    }
    if (threadIdx.x == 0) out[0] = red[0];
}

extern "C" void kernel_launch(void* const* d_in, const int* in_sizes, int n_in,
                              void* d_out, int out_size, void* d_ws, size_t ws_size,
                              hipStream_t stream) {
    const float* flow = (const float*)d_in[0];
    float* row_sums = (float*)d_ws;          // [8192]
    float* col_sums = row_sums + NMAT;       // [8192]
    float* out      = (float*)d_out;         // [1] f32

    zero_kernel<<<(2 * NMAT + 255) / 256, 256, 0, stream>>>(row_sums, 2 * NMAT);

    dim3 grid(NMAT / TILE_COLS, NMAT / ROWS_PER_BLOCK);   // 8 x 128 = 1024 blocks
    sums_kernel<<<grid, 256, 0, stream>>>(flow, row_sums, col_sums);

    finalize_kernel<<<1, 256, 0, stream>>>(row_sums, col_sums, out);
}